// MutualRec_21311627723303
// MI455X (gfx1250) — compile-verified
//
#include <hip/hip_runtime.h>
#include <hip/hip_bf16.h>
#include <cstdint>

// ---------------------------------------------------------------------------
// MutualRec forward on gfx1250.
//  * Dense [N,64]x[64,64] GEMMs (~16 FLOP/byte -> HBM-bound) run through
//    V_WMMA_F32_16X16X4_F32 (fp32 in/out, no precision loss, WMMA path).
//  * Edge segment-softmax via wave-per-edge kernels + float atomics
//    (ordered-int encoding for segment max).
//  * Workspace: 6 node buffers + edge buffers ~= 163 MB of d_ws.
// ---------------------------------------------------------------------------

#define NUSR 100000
#define NITM 100000
#define DIMV 64
#define NRE  1000000
#define NLE  500000
#define NEG_SLOPE 0.2f

typedef __attribute__((ext_vector_type(2))) float v2f;
typedef __attribute__((ext_vector_type(8))) float v8f;

__device__ __forceinline__ int f2ord(float f) {
  int i = __float_as_int(f);
  return (i >= 0) ? i : (i ^ 0x7fffffff);
}
__device__ __forceinline__ float ord2f(int i) {
  return __int_as_float((i >= 0) ? i : (i ^ 0x7fffffff));
}
#define ORD_NEG_INF ((int)0x807fffff) /* f2ord(-inf) */

// ---------------------------------------------------------------------------
// Fills
// ---------------------------------------------------------------------------
__global__ void fill_f32_kernel(float* __restrict__ p, float v, size_t n) {
  size_t i = (size_t)blockIdx.x * blockDim.x + threadIdx.x;
  if (i < n) p[i] = v;
}
__global__ void fill_i32_kernel(int* __restrict__ p, int v, size_t n) {
  size_t i = (size_t)blockIdx.x * blockDim.x + threadIdx.x;
  if (i < n) p[i] = v;
}

// ---------------------------------------------------------------------------
// WMMA GEMM: Y[N,64] = X[N,64] @ W[64,64] (+bias) (+Y if acc)
// Block: 256 threads = 8 waves; 32 rows x 64 cols per block.
// Wave w: row-tile = (w>>2)*16, col-tile = (w&3)*16. 16 WMMA steps over K.
// ---------------------------------------------------------------------------
__global__ __launch_bounds__(256) void gemm64_wmma_kernel(
    const float* __restrict__ X, const float* __restrict__ W,
    const float* __restrict__ bias, float* __restrict__ Y, int N, int acc) {
  __shared__ float Xs[32 * 66];  // padded stride 66 (even -> b64 LDS loads ok)
  __shared__ float Wls[64 * 66];

  const int tid = threadIdx.x;
  const int row0 = blockIdx.x * 32;

  // Stage 32x64 X tile (coalesced), zero-fill OOB rows.
#pragma unroll
  for (int i = 0; i < 8; ++i) {
    int li = tid + i * 256;
    int r = li >> 6, cc = li & 63;
    int gr = row0 + r;
    Xs[r * 66 + cc] = (gr < N) ? X[(size_t)gr * DIMV + cc] : 0.0f;
  }
  // Stage full 64x64 W.
#pragma unroll
  for (int i = 0; i < 16; ++i) {
    int li = tid + i * 256;
    int r = li >> 6, cc = li & 63;
    Wls[r * 66 + cc] = W[li];
  }
  __syncthreads();

  const int wave = tid >> 5;
  const int lane = tid & 31;
  const int rt = wave >> 2;      // 0..1
  const int ct = wave & 3;       // 0..3
  const int hi = lane >> 4;      // 0/1 (selects K pair per ISA A layout)
  const int l16 = lane & 15;
  const int col = ct * 16 + l16;

  v8f c;
  if (acc) {
#pragma unroll
    for (int v = 0; v < 8; ++v) {
      int gr = row0 + rt * 16 + v + 8 * hi;  // C/D layout: M = v + 8*hi
      c[v] = (gr < N) ? Y[(size_t)gr * DIMV + col] : 0.0f;
    }
  } else {
#pragma unroll
    for (int v = 0; v < 8; ++v) c[v] = 0.0f;
  }

#if __has_builtin(__builtin_amdgcn_wmma_f32_16x16x4_f32)
#pragma unroll
  for (int k = 0; k < DIMV; k += 4) {
    // A 16x4 f32: lanes0-15 K={k,k+1}, lanes16-31 K={k+2,k+3}; M = lane&15
    v2f a = *(const v2f*)&Xs[(rt * 16 + l16) * 66 + k + 2 * hi];
    // B 4x16 f32 (mirrored): N = lane&15, K pair selected by hi
    v2f b;
    b.x = Wls[(k + 2 * hi) * 66 + col];
    b.y = Wls[(k + 2 * hi + 1) * 66 + col];
    c = __builtin_amdgcn_wmma_f32_16x16x4_f32(false, a, false, b, (short)0, c,
                                              false, false);
  }
#else
  // Scalar fallback matching the same fragment layout (should not trigger).
#pragma unroll
  for (int v = 0; v < 8; ++v) {
    int r = rt * 16 + v + 8 * hi;
    float s = 0.0f;
    for (int k = 0; k < DIMV; ++k) s += Xs[r * 66 + k] * Wls[k * 66 + col];
    c[v] += s;
  }
#endif

  float bv = bias ? bias[col] : 0.0f;
#pragma unroll
  for (int v = 0; v < 8; ++v) {
    int gr = row0 + rt * 16 + v + 8 * hi;
    if (gr < N) Y[(size_t)gr * DIMV + col] = c[v] + bv;
  }
}

// ---------------------------------------------------------------------------
// GATv2 edge kernels (wave per edge, 2 components per lane)
// ---------------------------------------------------------------------------
__global__ __launch_bounds__(256) void edge_score_kernel(
    const float* __restrict__ fs, const float* __restrict__ fd,
    const float* __restrict__ attn, const int* __restrict__ src,
    const int* __restrict__ dst, int E, float* __restrict__ s_out,
    int* __restrict__ mseg) {
  int e = blockIdx.x * 8 + (threadIdx.x >> 5);
  if (e >= E) return;
  int lane = threadIdx.x & 31;
  int c0 = lane * 2;
  int sN = src[e], dN = dst[e];
  size_t so = (size_t)sN * DIMV, dofs = (size_t)dN * DIMV;
  float a0 = fs[so + c0] + fd[dofs + c0];
  float a1 = fs[so + c0 + 1] + fd[dofs + c0 + 1];
  a0 = (a0 > 0.0f) ? a0 : a0 * NEG_SLOPE;
  a1 = (a1 > 0.0f) ? a1 : a1 * NEG_SLOPE;
  float p = a0 * attn[c0] + a1 * attn[c0 + 1];
#pragma unroll
  for (int off = 16; off > 0; off >>= 1) p += __shfl_xor(p, off, 32);
  if (lane == 0) {
    s_out[e] = p;
    atomicMax(&mseg[dN], f2ord(p));
  }
}

__global__ void edge_exp_kernel(const float* __restrict__ s,
                                const int* __restrict__ mseg,
                                const int* __restrict__ dst, int E,
                                float* __restrict__ ex_out,
                                float* __restrict__ den) {
  int e = blockIdx.x * blockDim.x + threadIdx.x;
  if (e >= E) return;
  int d = dst[e];
  float ex = __expf(s[e] - ord2f(mseg[d]));
  ex_out[e] = ex;
  atomicAdd(&den[d], ex);
}

__global__ __launch_bounds__(256) void edge_aggr_kernel(
    const float* __restrict__ fs, const int* __restrict__ src,
    const int* __restrict__ dst, const float* __restrict__ ex,
    const float* __restrict__ den, int E, float* __restrict__ out) {
  int e = blockIdx.x * 8 + (threadIdx.x >> 5);
  if (e >= E) return;
  int lane = threadIdx.x & 31;
  int c0 = lane * 2;
  int sN = src[e], dN = dst[e];
  float alpha = ex[e] / den[dN];
  atomicAdd(&out[(size_t)dN * DIMV + c0], alpha * fs[(size_t)sN * DIMV + c0]);
  atomicAdd(&out[(size_t)dN * DIMV + c0 + 1],
            alpha * fs[(size_t)sN * DIMV + c0 + 1]);
}

// ---------------------------------------------------------------------------
// ChebConv helpers
// ---------------------------------------------------------------------------
__global__ void deg_count_kernel(const int* __restrict__ dst, int E,
                                 float* __restrict__ indeg) {
  int e = blockIdx.x * blockDim.x + threadIdx.x;
  if (e < E) atomicAdd(&indeg[dst[e]], 1.0f);
}

__global__ void make_dinv_kernel(float* __restrict__ d, int N) {
  int i = blockIdx.x * blockDim.x + threadIdx.x;
  if (i < N) d[i] = rsqrtf(fmaxf(d[i], 1.0f));
}

__global__ __launch_bounds__(256) void cheb_scatter_kernel(
    const float* __restrict__ x, const float* __restrict__ dinv,
    const int* __restrict__ src, const int* __restrict__ dst, int E,
    float* __restrict__ tmp) {
  int e = blockIdx.x * 8 + (threadIdx.x >> 5);
  if (e >= E) return;
  int lane = threadIdx.x & 31;
  int c0 = lane * 2;
  int sN = src[e], dN = dst[e];
  float w = dinv[sN];
  atomicAdd(&tmp[(size_t)dN * DIMV + c0], x[(size_t)sN * DIMV + c0] * w);
  atomicAdd(&tmp[(size_t)dN * DIMV + c0 + 1],
            x[(size_t)sN * DIMV + c0 + 1] * w);
}

__global__ void cheb_scale_kernel(float* __restrict__ t,
                                  const float* __restrict__ dinv, size_t n) {
  size_t i = (size_t)blockIdx.x * blockDim.x + threadIdx.x;
  if (i < n) t[i] *= dinv[i >> 6];
}

__global__ void cheb_t1_kernel(float* __restrict__ T1,
                               const float* __restrict__ h,
                               const float* __restrict__ T0,
                               const float* __restrict__ lam, size_t n) {
  size_t i = (size_t)blockIdx.x * blockDim.x + threadIdx.x;
  if (i >= n) return;
  float re = 2.0f / lam[0];
  T1[i] = -re * h[i] + T0[i] * (re - 1.0f);
}

__global__ void cheb_t2_kernel(float* __restrict__ T2,
                               const float* __restrict__ h,
                               const float* __restrict__ T1,
                               const float* __restrict__ T0,
                               const float* __restrict__ lam, size_t n) {
  size_t i = (size_t)blockIdx.x * blockDim.x + threadIdx.x;
  if (i >= n) return;
  float re = 2.0f / lam[0];
  T2[i] = -2.0f * re * h[i] + 2.0f * (re - 1.0f) * T1[i] - T0[i];
}

// ---------------------------------------------------------------------------
// Mutualistic layer: h_m = h_uP*h_uS; out = h_m * softmax(row) (wave per row)
// ---------------------------------------------------------------------------
__global__ __launch_bounds__(256) void mutual_kernel(
    const float* __restrict__ hP, const float* __restrict__ hS,
    float* __restrict__ mP, float* __restrict__ mS, int N) {
  int r = blockIdx.x * 8 + (threadIdx.x >> 5);
  if (r >= N) return;
  int lane = threadIdx.x & 31;
  int c0 = lane * 2;
  size_t base = (size_t)r * DIMV;
  float p0 = hP[base + c0], p1 = hP[base + c0 + 1];
  float s0 = hS[base + c0], s1 = hS[base + c0 + 1];
  float xm = fmaxf(p0, p1), ym = fmaxf(s0, s1);
#pragma unroll
  for (int off = 16; off > 0; off >>= 1) {
    xm = fmaxf(xm, __shfl_xor(xm, off, 32));
    ym = fmaxf(ym, __shfl_xor(ym, off, 32));
  }
  float ep0 = __expf(p0 - xm), ep1 = __expf(p1 - xm);
  float es0 = __expf(s0 - ym), es1 = __expf(s1 - ym);
  float sp = ep0 + ep1, ss = es0 + es1;
#pragma unroll
  for (int off = 16; off > 0; off >>= 1) {
    sp += __shfl_xor(sp, off, 32);
    ss += __shfl_xor(ss, off, 32);
  }
  float m0 = p0 * s0, m1 = p1 * s1;
  mP[base + c0] = m0 * (ep0 / sp);
  mP[base + c0 + 1] = m1 * (ep1 / sp);
  mS[base + c0] = m0 * (es0 / ss);
  mS[base + c0 + 1] = m1 * (es1 / ss);
}

// ---------------------------------------------------------------------------
// Per-edge dot product (wave per edge)
// ---------------------------------------------------------------------------
__global__ __launch_bounds__(256) void edge_dot_kernel(
    const float* __restrict__ x, const float* __restrict__ y,
    const int* __restrict__ src, const int* __restrict__ dst, int E,
    float* __restrict__ out) {
  int e = blockIdx.x * 8 + (threadIdx.x >> 5);
  if (e >= E) return;
  int lane = threadIdx.x & 31;
  int c0 = lane * 2;
  size_t so = (size_t)src[e] * DIMV, dofs = (size_t)dst[e] * DIMV;
  float p = x[so + c0] * y[dofs + c0] + x[so + c0 + 1] * y[dofs + c0 + 1];
#pragma unroll
  for (int off = 16; off > 0; off >>= 1) p += __shfl_xor(p, off, 32);
  if (lane == 0) out[e] = p;
}

// ---------------------------------------------------------------------------
// Host-side helpers
// ---------------------------------------------------------------------------
static inline void h_fill_f32(float* p, float v, size_t n, hipStream_t st) {
  fill_f32_kernel<<<(unsigned)((n + 255) / 256), 256, 0, st>>>(p, v, n);
}
static inline void h_fill_i32(int* p, int v, size_t n, hipStream_t st) {
  fill_i32_kernel<<<(unsigned)((n + 255) / 256), 256, 0, st>>>(p, v, n);
}
static inline void h_gemm(const float* X, const float* W, const float* bias,
                          float* Y, int N, int acc, hipStream_t st) {
  gemm64_wmma_kernel<<<(N + 31) / 32, 256, 0, st>>>(X, W, bias, Y, N, acc);
}

static void run_gat(const float* Wsrc, const float* bsrc, const float* Wdst,
                    const float* bdst, const float* attn, const float* h_src,
                    int Ns, const float* h_dst, int Nd, const int* src,
                    const int* dst, int E, float* fsb, float* fdb, float* outb,
                    float* sbuf, float* exbuf, int* mbuf, float* den,
                    hipStream_t st) {
  h_gemm(h_src, Wsrc, bsrc, fsb, Ns, 0, st);
  h_gemm(h_dst, Wdst, bdst, fdb, Nd, 0, st);
  h_fill_i32(mbuf, ORD_NEG_INF, (size_t)Nd, st);
  h_fill_f32(den, 0.0f, (size_t)Nd, st);
  h_fill_f32(outb, 0.0f, (size_t)Nd * DIMV, st);
  edge_score_kernel<<<(E + 7) / 8, 256, 0, st>>>(fsb, fdb, attn, src, dst, E,
                                                 sbuf, mbuf);
  edge_exp_kernel<<<(E + 255) / 256, 256, 0, st>>>(sbuf, mbuf, dst, E, exbuf,
                                                   den);
  edge_aggr_kernel<<<(E + 7) / 8, 256, 0, st>>>(fsb, src, dst, exbuf, den, E,
                                                outb);
}

extern "C" void kernel_launch(void* const* d_in, const int* in_sizes, int n_in,
                              void* d_out, int out_size, void* d_ws,
                              size_t ws_size, hipStream_t stream) {
  (void)in_sizes; (void)n_in; (void)out_size; (void)ws_size;
  const float* user_emb = (const float*)d_in[0];
  const float* item_emb = (const float*)d_in[1];
  const int* rate_src = (const int*)d_in[2];
  const int* rate_dst = (const int*)d_in[3];
  const int* link_src = (const int*)d_in[4];
  const int* link_dst = (const int*)d_in[5];
  const int* neg_rate_src = (const int*)d_in[6];
  const int* neg_rate_dst = (const int*)d_in[7];
  const int* neg_link_src = (const int*)d_in[8];
  const int* neg_link_dst = (const int*)d_in[9];
  const float* lambda_max = (const float*)d_in[10];
  const float* gat_Wsrc = (const float*)d_in[11];
  const float* gat_bsrc = (const float*)d_in[12];
  const float* gat_Wdst = (const float*)d_in[13];
  const float* gat_bdst = (const float*)d_in[14];
  const float* gat_attn = (const float*)d_in[15];
  const float* W_out = (const float*)d_in[16];
  const float* b_out = (const float*)d_in[17];
  const float* cheb_W = (const float*)d_in[18];
  const float* cheb_b = (const float*)d_in[19];
  const float* Wc = (const float*)d_in[20];
  const float* bc = (const float*)d_in[21];
  const float* Wsm = (const float*)d_in[22];
  const float* bsm = (const float*)d_in[23];
  const float* WpP = (const float*)d_in[24];
  const float* bpP = (const float*)d_in[25];
  const float* WpS = (const float*)d_in[26];
  const float* bpS = (const float*)d_in[27];
  float* out = (float*)d_out;

  const size_t NODE = (size_t)NUSR * DIMV;  // 6.4M floats (NUSR == NITM)
  float* A = (float*)d_ws;
  float* B = A + NODE;
  float* C = B + NODE;
  float* Dh = C + NODE;
  float* Eh = Dh + NODE;
  float* Fh = Eh + NODE;
  float* sbuf = Fh + NODE;           // NRE floats
  float* exbuf = sbuf + NRE;         // NRE floats
  int* mbuf = (int*)(exbuf + NRE);   // NUSR ints
  float* den = (float*)(mbuf + NUSR);
  float* dinv = den + NUSR;
  const int WW = DIMV * DIMV;  // 4096

#define GATW(i) gat_Wsrc + (i)*WW, gat_bsrc + (i)*DIMV, gat_Wdst + (i)*WW, \
                gat_bdst + (i)*DIMV, gat_attn + (i)*DIMV

  // ---- SpatialAttentionLayer_GAT ----
  // h1_item = gat0(user->item, rate)           -> C
  run_gat(GATW(0), user_emb, NUSR, item_emb, NITM, rate_src, rate_dst, NRE, A,
          B, C, sbuf, exbuf, mbuf, den, stream);
  // h2_user = gat1(item->user, reversed rate)  -> Dh
  run_gat(GATW(1), item_emb, NITM, user_emb, NUSR, rate_dst, rate_src, NRE, A,
          B, Dh, sbuf, exbuf, mbuf, den, stream);
  // item_infl = gat2(h1_item->user, rev rate)  -> Eh   (C dead after)
  run_gat(GATW(2), C, NITM, user_emb, NUSR, rate_dst, rate_src, NRE, A, B, Eh,
          sbuf, exbuf, mbuf, den, stream);
  // social_item = gat3(h2_user->user, link)    -> C    (Dh dead after)
  run_gat(GATW(3), Dh, NUSR, user_emb, NUSR, link_src, link_dst, NLE, A, B, C,
          sbuf, exbuf, mbuf, den, stream);
  // user_pref = concat(item_infl, social_item) @ W_out + b_out -> Dh
  h_gemm(Eh, W_out, nullptr, Dh, NUSR, 0, stream);
  h_gemm(C, W_out + WW, b_out, Dh, NUSR, 1, stream);

  // ---- SpectralAttentionLayer: ChebConv(k=3) + GATv2 ----
  h_fill_f32(dinv, 0.0f, NUSR, stream);
  deg_count_kernel<<<(NLE + 255) / 256, 256, 0, stream>>>(link_dst, NLE, dinv);
  make_dinv_kernel<<<(NUSR + 255) / 256, 256, 0, stream>>>(dinv, NUSR);
  // rst = T0 @ W0 -> Eh
  h_gemm(user_emb, cheb_W, nullptr, Eh, NUSR, 0, stream);
  // h = ahat(T0) -> A
  h_fill_f32(A, 0.0f, NODE, stream);
  cheb_scatter_kernel<<<(NLE + 7) / 8, 256, 0, stream>>>(user_emb, dinv,
                                                         link_src, link_dst,
                                                         NLE, A);
  cheb_scale_kernel<<<(unsigned)((NODE + 255) / 256), 256, 0, stream>>>(A, dinv,
                                                                        NODE);
  // T1 -> B ; rst += T1 @ W1
  cheb_t1_kernel<<<(unsigned)((NODE + 255) / 256), 256, 0, stream>>>(
      B, A, user_emb, lambda_max, NODE);
  h_gemm(B, cheb_W + WW, nullptr, Eh, NUSR, 1, stream);
  // h = ahat(T1) -> A
  h_fill_f32(A, 0.0f, NODE, stream);
  cheb_scatter_kernel<<<(NLE + 7) / 8, 256, 0, stream>>>(B, dinv, link_src,
                                                         link_dst, NLE, A);
  cheb_scale_kernel<<<(unsigned)((NODE + 255) / 256), 256, 0, stream>>>(A, dinv,
                                                                        NODE);
  // T2 -> C ; rst += T2 @ W2 + cheb_b
  cheb_t2_kernel<<<(unsigned)((NODE + 255) / 256), 256, 0, stream>>>(
      C, A, B, user_emb, lambda_max, NODE);
  h_gemm(C, cheb_W + 2 * WW, cheb_b, Eh, NUSR, 1, stream);
  // user_social = gat4(rst->rst, link) -> C   (Eh dead after)
  run_gat(GATW(4), Eh, NUSR, Eh, NUSR, link_src, link_dst, NLE, A, B, C, sbuf,
          exbuf, mbuf, den, stream);

  // ---- MutualisicLayer ----
  // h_uP = concat(user_pref, user_emb) @ Wc + bc -> Eh  (Dh consumed)
  h_gemm(Dh, Wc, nullptr, Eh, NUSR, 0, stream);
  h_gemm(user_emb, Wc + WW, bc, Eh, NUSR, 1, stream);
  // h_uS = concat(user_social, user_emb) @ Ws + bs -> Dh (C consumed)
  h_gemm(C, Wsm, nullptr, Dh, NUSR, 0, stream);
  h_gemm(user_emb, Wsm + WW, bsm, Dh, NUSR, 1, stream);
  // h_mP -> A, h_mS -> B
  mutual_kernel<<<(NUSR + 7) / 8, 256, 0, stream>>>(Eh, Dh, A, B, NUSR);

  // ---- PredictionLayer ----
  // h_new_P = concat(h_mP, h_uP) @ WpP + bpP -> C
  h_gemm(A, WpP, nullptr, C, NUSR, 0, stream);
  h_gemm(Eh, WpP + WW, bpP, C, NUSR, 1, stream);
  // h_new_S = concat(h_mS, h_uS) @ WpS + bpS -> Fh
  h_gemm(B, WpS, nullptr, Fh, NUSR, 0, stream);
  h_gemm(Dh, WpS + WW, bpS, Fh, NUSR, 1, stream);

  // ---- HeteroDotProductPredictor ----
  edge_dot_kernel<<<(NRE + 7) / 8, 256, 0, stream>>>(C, item_emb, rate_src,
                                                     rate_dst, NRE, out);
  edge_dot_kernel<<<(NRE + 7) / 8, 256, 0, stream>>>(
      C, item_emb, neg_rate_src, neg_rate_dst, NRE, out + NRE);
  edge_dot_kernel<<<(NLE + 7) / 8, 256, 0, stream>>>(
      Fh, user_emb, link_src, link_dst, NLE, out + 2 * (size_t)NRE);
  edge_dot_kernel<<<(NLE + 7) / 8, 256, 0, stream>>>(
      Fh, user_emb, neg_link_src, neg_link_dst, NLE,
      out + 2 * (size_t)NRE + NLE);
#undef GATW
}